// SSIMLoss_4690104287880
// MI455X (gfx1250) — compile-verified
//
#include <hip/hip_runtime.h>
#include <stdint.h>

// ---------------------------------------------------------------------------
// SSIM loss (7x7 box, VALID) for B=32, 1x1024x1024 f32, scalar output.
//
// Memory-bound problem (AI ~= 5 FLOP/byte): read x,y once (268 MB -> ~11.5us
// floor at 23.3 TB/s). Strategy:
//   * TDM (tensor_load_to_lds) stages 24x24 f32 tiles (16x16 outputs + halo)
//     into per-wave LDS regions; hardware zero-fills out-of-tensor reads at
//     the right/bottom edges. Wait with s_wait_tensorcnt.
//   * 7x7 box sums as two banded f32 WMMA matmuls on the matrix pipe
//     (V = S*M vertical, O = V*T horizontal; S,T are 0/1 bands generated
//     from lane IDs). f32 end-to-end preserves the uxx-ux^2 cancellation.
//   * Pointwise SSIM per lane (8 outputs each, WMMA C/D layout), masked
//     wave/workgroup reduction, one f32 per WG into d_ws (write, no atomics),
//     tiny finalize kernel -> 1 - mean.
// ---------------------------------------------------------------------------

typedef __attribute__((ext_vector_type(2))) float v2f;
typedef __attribute__((ext_vector_type(8))) float v8f;
typedef __attribute__((ext_vector_type(4))) unsigned int v4u;
typedef __attribute__((ext_vector_type(4))) int v4i;
typedef __attribute__((ext_vector_type(8))) int v8i;

#define IMG_H 1024
#define IMG_W 1024
#define OUT_D 1018          // 1024 - 6 (VALID, win=7)
#define TILES_X 64          // ceil(1018/16)
#define TILES_Y 64
#define NBATCH 32
#define WAVES 8             // waves per workgroup, 1 tile each
#define TW 24               // LDS tile: 24 rows x 24 cols f32 (16 + 6 halo, padded)

static __device__ __forceinline__ v8f wmma4(v2f a, v2f b, v8f c) {
  // V_WMMA_F32_16X16X4_F32 : D = A(16x4) * B(4x16) + C(16x16)
  return __builtin_amdgcn_wmma_f32_16x16x4_f32(
      /*neg_a=*/false, a, /*neg_b=*/false, b,
      /*c_mod=*/(short)0, c, /*reuse_a=*/false, /*reuse_b=*/false);
}

// TDM 2D tile load: 24x24 f32 tile at gsrc (row stride IMG_W) -> LDS offset.
// rem_cols/rem_rows = elements remaining to the tensor edge from the tile
// origin; reads past them return zero (hardware OOB fill).
static __device__ __forceinline__ void tdm_load_tile(unsigned lds_off,
                                                     const float* gsrc,
                                                     int rem_cols, int rem_rows) {
  unsigned long long ga = (unsigned long long)(uintptr_t)gsrc;
  v4u g0;
  g0[0] = 1u;                                           // count=1, user mode
  g0[1] = lds_off;                                      // lds_addr (bytes)
  g0[2] = (unsigned)ga;                                 // global_addr[31:0]
  g0[3] = (unsigned)((ga >> 32) & 0x1FFFFFFu)           // global_addr[56:32]
          | (2u << 30);                                 // type=2 ("image")
  v8i g1;
  g1[0] = (int)(2u << 16);                              // data_size=4B; mask=0
  g1[1] = (int)(((unsigned)rem_cols & 0xFFFFu) << 16);  // tensor_dim0[15:0]
  g1[2] = (int)((((unsigned)rem_cols >> 16) & 0xFFFFu)
          | (((unsigned)rem_rows & 0xFFFFu) << 16));    // td0 hi | td1 lo
  g1[3] = (int)((((unsigned)rem_rows >> 16) & 0xFFFFu)
          | ((unsigned)TW << 16));                      // td1 hi | tile_dim0=24
  g1[4] = (int)TW;                                      // tile_dim1=24, tile_dim2=0
  g1[5] = IMG_W;                                        // tensor_dim0_stride lo
  g1[6] = 0;                                            // stride hi / dim1_stride lo
  g1[7] = 0;
  v4i z4 = {0, 0, 0, 0};
  v8i z8 = {0, 0, 0, 0, 0, 0, 0, 0};
  // amdgpu-toolchain (clang-23) 6-arg form: (g0, g1, g2, g3, g4, cpol)
  __builtin_amdgcn_tensor_load_to_lds(g0, g1, z4, z4, z8, 0);
}

__launch_bounds__(256)
__global__ void ssim_tiles(const float* __restrict__ x, const float* __restrict__ y,
                           const float* __restrict__ drp, float* __restrict__ partial) {
  __shared__ float lx[WAVES][TW * TW];
  __shared__ float ly[WAVES][TW * TW];
  __shared__ float lv[WAVES][16 * TW];   // per-wave vertical-sum relayout scratch
  __shared__ float red[WAVES];

  const int wv = threadIdx.x >> 5;
  const int lane = threadIdx.x & 31;
  const int half = lane >> 4;       // 0: lanes 0-15, 1: lanes 16-31
  const int lr = lane & 15;

  const int tc = blockIdx.x * WAVES + wv;   // tile col [0,64)
  const int tr = blockIdx.y;                // tile row [0,64)
  const int b = blockIdx.z;
  const int r0 = tr * 16, c0 = tc * 16;

  const float* xin = x + ((size_t)b * IMG_H + r0) * IMG_W + c0;
  const float* yin = y + ((size_t)b * IMG_H + r0) * IMG_W + c0;

  // Async DMA both tiles into this wave's LDS regions, then wait.
  tdm_load_tile((unsigned)(uintptr_t)&lx[wv][0], xin, IMG_W - c0, IMG_H - r0);
  tdm_load_tile((unsigned)(uintptr_t)&ly[wv][0], yin, IMG_W - c0, IMG_H - r0);
  __builtin_amdgcn_s_wait_tensorcnt(0);

  const float* ltx = &lx[wv][0];
  const float* lty = &ly[wv][0];
  float* lvm = &lv[wv][0];

  v8f outm[5];   // raw 7x7 sums of x, y, xx, yy, xy (16x16 C/D layout)

#pragma unroll
  for (int m = 0; m < 5; ++m) {
    // ---- pass 1: vertical band  V(16 x 24cols) = S(16x24) * M(24x24) ----
#pragma unroll
    for (int cb = 0; cb < 2; ++cb) {
      v8f acc = {0.f, 0.f, 0.f, 0.f, 0.f, 0.f, 0.f, 0.f};
      int col = cb * 16 + lr;
      if (col > TW - 1) col = TW - 1;   // clamp; cols >= 22 are band-killed later
#pragma unroll
      for (int t = 0; t < 6; ++t) {
        const int k0 = 4 * t + half * 2;
        const int k1 = k0 + 1;
        // A = S band chunk: S[i][k] = 1 iff i <= k <= i+6  (i = lr)
        v2f a;
        a.x = (k0 >= lr && k0 <= lr + 6) ? 1.f : 0.f;
        a.y = (k1 >= lr && k1 <= lr + 6) ? 1.f : 0.f;
        // B = data rows k0/k1 of map m (B layout: v0 lanes0-15=K0,16-31=K2; v1=K1,K3)
        const float x0 = ltx[k0 * TW + col], x1 = ltx[k1 * TW + col];
        const float y0 = lty[k0 * TW + col], y1 = lty[k1 * TW + col];
        v2f bb;
        bb.x = (m == 0) ? x0 : (m == 1) ? y0 : (m == 2) ? x0 * x0
               : (m == 3) ? y0 * y0 : x0 * y0;
        bb.y = (m == 0) ? x1 : (m == 1) ? y1 : (m == 2) ? x1 * x1
               : (m == 3) ? y1 * y1 : x1 * y1;
        acc = wmma4(a, bb, acc);
      }
      // Spill V block to LDS in row-major (C/D layout: vgpr v -> row v + half*8)
      const int scol = cb * 16 + lr;
      if (scol < TW) {
#pragma unroll
        for (int v = 0; v < 8; ++v)
          lvm[(v + half * 8) * TW + scol] = acc[v];
      }
    }
    // ---- pass 2: horizontal band  O(16x16) = V(16x24) * T(24x16) ----
    v8f o = {0.f, 0.f, 0.f, 0.f, 0.f, 0.f, 0.f, 0.f};
#pragma unroll
    for (int t = 0; t < 6; ++t) {
      const int k0 = 4 * t + half * 2;
      const int k1 = k0 + 1;
      v2f a;   // A = V rows (row = lr), cols k0/k1
      a.x = lvm[lr * TW + k0];
      a.y = lvm[lr * TW + k1];
      v2f bb;  // B = T band: T[k][n] = 1 iff n <= k <= n+6  (n = lr)
      bb.x = (lr <= k0 && k0 <= lr + 6) ? 1.f : 0.f;
      bb.y = (lr <= k1 && k1 <= lr + 6) ? 1.f : 0.f;
      o = wmma4(a, bb, o);
    }
    outm[m] = o;
  }

  // ---- pointwise SSIM + masked reduction ----
  const float dr = drp[0];
  const float c1v = (0.01f * dr) * (0.01f * dr);
  const float c2v = (0.03f * dr) * (0.03f * dr);
  const float inv49 = 1.0f / 49.0f;
  const float covn = 49.0f / 48.0f;

  float lsum = 0.f;
#pragma unroll
  for (int v = 0; v < 8; ++v) {
    const int orow = r0 + v + half * 8;
    const int ocol = c0 + lr;
    const float sx = outm[0][v] * inv49;
    const float sy = outm[1][v] * inv49;
    const float sxx = outm[2][v] * inv49;
    const float syy = outm[3][v] * inv49;
    const float sxy = outm[4][v] * inv49;
    const float vx = covn * (sxx - sx * sx);
    const float vy = covn * (syy - sy * sy);
    const float vxy = covn * (sxy - sx * sy);
    const float num = (2.f * sx * sy + c1v) * (2.f * vxy + c2v);
    const float den = (sx * sx + sy * sy + c1v) * (vx + vy + c2v);
    const float s = num / den;
    if (orow < OUT_D && ocol < OUT_D) lsum += s;
  }
#pragma unroll
  for (int off = 16; off > 0; off >>= 1) lsum += __shfl_xor(lsum, off, 32);
  if (lane == 0) red[wv] = lsum;
  __syncthreads();
  if (threadIdx.x == 0) {
    float s = 0.f;
#pragma unroll
    for (int i = 0; i < WAVES; ++i) s += red[i];
    partial[blockIdx.x + (size_t)gridDim.x * (blockIdx.y + (size_t)gridDim.y * blockIdx.z)] = s;
  }
}

__launch_bounds__(256)
__global__ void ssim_finalize(const float* __restrict__ partial, float* __restrict__ out, int n) {
  __shared__ float sm[256];
  float s = 0.f;
  for (int i = threadIdx.x; i < n; i += 256) s += partial[i];
  sm[threadIdx.x] = s;
  __syncthreads();
  for (int w = 128; w > 0; w >>= 1) {
    if ((int)threadIdx.x < w) sm[threadIdx.x] += sm[threadIdx.x + w];
    __syncthreads();
  }
  if (threadIdx.x == 0) {
    const float N = (float)NBATCH * (float)OUT_D * (float)OUT_D;
    out[0] = 1.0f - sm[0] / N;
  }
}

extern "C" void kernel_launch(void* const* d_in, const int* in_sizes, int n_in,
                              void* d_out, int out_size, void* d_ws, size_t ws_size,
                              hipStream_t stream) {
  const float* x = (const float*)d_in[0];
  const float* y = (const float*)d_in[1];
  const float* dr = (const float*)d_in[2];
  float* partial = (float*)d_ws;   // 16384 f32 = 64 KB, fully rewritten each call

  dim3 grid(TILES_X / WAVES, TILES_Y, NBATCH);   // (8, 64, 32)
  ssim_tiles<<<grid, 256, 0, stream>>>(x, y, dr, partial);

  const int nPart = (TILES_X / WAVES) * TILES_Y * NBATCH;
  ssim_finalize<<<1, 256, 0, stream>>>(partial, (float*)d_out, nPart);
}